// Group_11330123727109
// MI455X (gfx1250) — compile-verified
//
#include <hip/hip_runtime.h>
#include <hip/hip_bf16.h>
#include <stdint.h>

// Problem constants (from reference)
#define BATCH 32
#define NPTS  8192
#define NGRP  512
#define KNN_K 32
#define INF_F 1e10f

// FPS kernel config
#define FPS_THREADS 1024
#define FPS_PER_THR (NPTS / FPS_THREADS)   // 8

// KNN kernel config
#define KNN_THREADS 512                    // 16 waves
#define CENT_PER_WG 16
#define PTS_PER_LANE (NPTS / 32)           // 256

typedef __attribute__((ext_vector_type(2))) float v2f;
typedef __attribute__((ext_vector_type(8))) float v8f;

// ---------------------------------------------------------------------------
// Kernel 1: farthest point sampling. One workgroup per batch (32 WGs, 1024
// threads). All 8192 points live in registers (8/thread). Per step:
//   dist[j] = min(dist[j], |p-c|^2); argmax via packed u64 key
//   key = (f32bits(dist) << 32) | (0xFFFFFFFF - idx)   (max => first-occurrence tie)
// Also emits center / norm_center / idx_new (the reference records the carry
// *before* each step, so idx_new[b][0] == 0).
// ---------------------------------------------------------------------------
__global__ __launch_bounds__(FPS_THREADS)
void fps_kernel(const float* __restrict__ xyz, const float* __restrict__ nrm,
                int* __restrict__ idx_new, float* __restrict__ center,
                float* __restrict__ norm_center)
{
    const int b   = blockIdx.x;
    const int tid = threadIdx.x;
    const float* xb = xyz + (size_t)b * NPTS * 3;
    const float* nb = nrm + (size_t)b * NPTS * 3;

    __shared__ unsigned s_whi[32], s_wlo[32];
    __shared__ unsigned s_bhi, s_blo;
    __shared__ float s_cx, s_cy, s_cz;

    float px[FPS_PER_THR], py[FPS_PER_THR], pz[FPS_PER_THR], dist[FPS_PER_THR];
#pragma unroll
    for (int j = 0; j < FPS_PER_THR; ++j) {
        const int n = j * FPS_THREADS + tid;
        px[j] = xb[n * 3 + 0];
        py[j] = xb[n * 3 + 1];
        pz[j] = xb[n * 3 + 2];
        dist[j] = INF_F;
    }

    int far = 0;
    for (int s = 0; s < NGRP; ++s) {
        // Owner of `far` records outputs and broadcasts centroid coords.
        if ((far & (FPS_THREADS - 1)) == tid) {
            const int j = far >> 10;           // far / 1024
            s_cx = px[j]; s_cy = py[j]; s_cz = pz[j];
            const size_t o = ((size_t)b * NGRP + s) * 3;
            center[o + 0] = px[j];
            center[o + 1] = py[j];
            center[o + 2] = pz[j];
            norm_center[o + 0] = nb[far * 3 + 0];
            norm_center[o + 1] = nb[far * 3 + 1];
            norm_center[o + 2] = nb[far * 3 + 2];
            idx_new[b * NGRP + s] = far;
        }
        __syncthreads();
        const float cx = s_cx, cy = s_cy, cz = s_cz;

        unsigned khi = 0u, klo = 0u;
#pragma unroll
        for (int j = 0; j < FPS_PER_THR; ++j) {
            const float dx = px[j] - cx, dy = py[j] - cy, dz = pz[j] - cz;
            const float d  = dx * dx + dy * dy + dz * dz;
            const float dm = fminf(dist[j], d);
            dist[j] = dm;
            const int n = j * FPS_THREADS + tid;
            const unsigned hi = __float_as_uint(dm);          // dm >= 0 -> monotone bits
            const unsigned lo = 0xFFFFFFFFu - (unsigned)n;    // smaller idx wins ties
            if (hi > khi || (hi == khi && lo > klo)) { khi = hi; klo = lo; }
        }
        // wave32 max-reduce
#pragma unroll
        for (int off = 16; off >= 1; off >>= 1) {
            const unsigned ohi = __shfl_xor(khi, off, 32);
            const unsigned olo = __shfl_xor(klo, off, 32);
            if (ohi > khi || (ohi == khi && olo > klo)) { khi = ohi; klo = olo; }
        }
        if ((tid & 31) == 0) { s_whi[tid >> 5] = khi; s_wlo[tid >> 5] = klo; }
        __syncthreads();
        if (tid < 32) {
            unsigned hi = s_whi[tid], lo = s_wlo[tid];
#pragma unroll
            for (int off = 16; off >= 1; off >>= 1) {
                const unsigned ohi = __shfl_xor(hi, off, 32);
                const unsigned olo = __shfl_xor(lo, off, 32);
                if (ohi > hi || (ohi == hi && olo > lo)) { hi = ohi; lo = olo; }
            }
            if (tid == 0) { s_bhi = hi; s_blo = lo; }
        }
        __syncthreads();
        far = (int)(0xFFFFFFFFu - s_blo);
    }
}

// ---------------------------------------------------------------------------
// Kernel 2: KNN + grouping. One WG of 512 threads (16 waves) per
// (batch, 16-centroid tile) -> 1024 WGs.
// Phase 1: V_WMMA_F32_16X16X4_F32 computes the [16 cent x 16 pts] dot tiles
//          (einsum 'bgc,bnc->bgn', K=3 padded to 4); d2 = |c|^2 - 2*dot + |p|^2
//          cached as f16 in 256 KB of LDS.
// Phase 2: each wave owns one centroid; 32 extraction rounds of a packed-key
//          wave argmin (ascending distance == top_k ordering); only the lane
//          that owned the extracted entry rescans its 256 values.
// Phase 3: lane r holds the r-th neighbor -> direct gather of outputs.
// ---------------------------------------------------------------------------
__global__ __launch_bounds__(KNN_THREADS)
void knn_group_kernel(const float* __restrict__ xyz, const float* __restrict__ nrm,
                      const float* __restrict__ center,
                      float* __restrict__ neighborhood, float* __restrict__ norm_grouped,
                      int* __restrict__ idx_out)
{
    extern __shared__ char smem[];
    _Float16* d2h  = (_Float16*)smem;                              // [16][8192]
    float*    s_sq = (float*)(smem + CENT_PER_WG * NPTS * 2);      // [16]
    float*    s_c  = s_sq + CENT_PER_WG;                           // [16][3]

    const int b    = blockIdx.x >> 5;          // 32 tiles per batch
    const int gt   = blockIdx.x & 31;
    const int g0   = gt * CENT_PER_WG;
    const int tid  = threadIdx.x;
    const int lane = tid & 31;
    const int wave = tid >> 5;

    const float* xb = xyz + (size_t)b * NPTS * 3;
    const float* nb = nrm + (size_t)b * NPTS * 3;
    const float* cb = center + ((size_t)b * NGRP + g0) * 3;

    if (tid < CENT_PER_WG) {
        const float cx = cb[tid * 3 + 0], cy = cb[tid * 3 + 1], cz = cb[tid * 3 + 2];
        s_c[tid * 3 + 0] = cx; s_c[tid * 3 + 1] = cy; s_c[tid * 3 + 2] = cz;
        s_sq[tid] = cx * cx + cy * cy + cz * cz;
    }
    __syncthreads();

    // A fragment: 16x4 f32 (ISA layout: lanes 0-15 hold K=0/1, lanes 16-31 K=2/3)
    const int m = lane & 15;
    v2f afrag;
    if (lane < 16) { afrag.x = s_c[m * 3 + 0]; afrag.y = s_c[m * 3 + 1]; }
    else           { afrag.x = s_c[m * 3 + 2]; afrag.y = 0.0f; }

    // ---- Phase 1: distance tiles via WMMA -------------------------------
    for (int t = wave; t < NPTS / 16; t += 16) {
        if (t + 16 < NPTS / 16)
            __builtin_prefetch(xb + (size_t)(t + 16) * 16 * 3, 0, 0);   // global_prefetch_b8

        const int p  = t * 16 + (lane & 15);
        const float x = xb[p * 3 + 0];
        const float y = xb[p * 3 + 1];
        const float z = xb[p * 3 + 2];
        const float psq = x * x + y * y + z * z;

        v2f bfrag;   // B: 4x16 f32, row-striped across lanes
        if (lane < 16) { bfrag.x = x; bfrag.y = y; }
        else           { bfrag.x = z; bfrag.y = 0.0f; }

        v8f acc = {};
        acc = __builtin_amdgcn_wmma_f32_16x16x4_f32(
            /*neg_a=*/false, afrag, /*neg_b=*/false, bfrag,
            /*c_mod=*/(short)0, acc, /*reuse_a=*/false, /*reuse_b=*/false);

        const int mbase = (lane < 16) ? 0 : 8;   // C/D layout: VGPR v -> row m = v (+8)
#pragma unroll
        for (int v = 0; v < 8; ++v) {
            const int mm = mbase + v;
            const float d2 = s_sq[mm] - 2.0f * acc[v] + psq;
            d2h[mm * NPTS + p] = (_Float16)d2;
        }
    }
    __syncthreads();

    // ---- Phase 2: per-wave top-32 extraction ----------------------------
    _Float16* row  = d2h + wave * NPTS;
    const int base = lane * PTS_PER_LANE;

    float lmin = 3.0e38f; int lpos = base;
    for (int q = 0; q < PTS_PER_LANE; ++q) {
        const float d = (float)row[base + q];
        if (d < lmin) { lmin = d; lpos = base + q; }
    }

    int   my_idx  = 0;
    for (int r = 0; r < KNN_K; ++r) {
        unsigned hi = __float_as_uint(lmin);   // d2 >= 0 in exact math; min-key
        unsigned lo = (unsigned)lpos;          // smaller point idx wins ties
#pragma unroll
        for (int off = 16; off >= 1; off >>= 1) {
            const unsigned ohi = __shfl_xor(hi, off, 32);
            const unsigned olo = __shfl_xor(lo, off, 32);
            if (ohi < hi || (ohi == hi && olo < lo)) { hi = ohi; lo = olo; }
        }
        const int wpos = (int)lo;
        if (lane == r) my_idx = wpos;
        if ((wpos >> 8) == lane) {             // owner: evict + rescan its 256 entries
            row[wpos] = (_Float16)65504.0f;
            lmin = 3.0e38f; lpos = base;
            for (int q = 0; q < PTS_PER_LANE; ++q) {
                const float d = (float)row[base + q];
                if (d < lmin) { lmin = d; lpos = base + q; }
            }
        }
    }

    // ---- Phase 3: gather outputs (lane r == neighbor rank r) ------------
    const int g = g0 + wave;
    const size_t o = (((size_t)b * NGRP + g) * KNN_K + lane);
    const int pi = my_idx;
    neighborhood[o * 3 + 0] = xb[pi * 3 + 0] - s_c[wave * 3 + 0];
    neighborhood[o * 3 + 1] = xb[pi * 3 + 1] - s_c[wave * 3 + 1];
    neighborhood[o * 3 + 2] = xb[pi * 3 + 2] - s_c[wave * 3 + 2];
    norm_grouped[o * 3 + 0] = nb[pi * 3 + 0];
    norm_grouped[o * 3 + 1] = nb[pi * 3 + 1];
    norm_grouped[o * 3 + 2] = nb[pi * 3 + 2];
    idx_out[o] = pi;
}

// ---------------------------------------------------------------------------
// Output layout (flat, reference return order):
//   neighborhood [32,512,32,3]  f32   1,572,864
//   center       [32,512,3]     f32      49,152
//   norm_grouped [32,512,32,3]  f32   1,572,864
//   norm_center  [32,512,3]     f32      49,152
//   idx          [32,512,32]    i32     524,288
//   idx_new      [32,512]       i32      16,384
// ---------------------------------------------------------------------------
extern "C" void kernel_launch(void* const* d_in, const int* in_sizes, int n_in,
                              void* d_out, int out_size, void* d_ws, size_t ws_size,
                              hipStream_t stream)
{
    const float* xyz = (const float*)d_in[0];
    const float* nrm = (const float*)d_in[1];

    float* out = (float*)d_out;
    float* neighborhood = out;
    float* center       = out + 1572864;
    float* norm_grouped = out + 1572864 + 49152;
    float* norm_center  = out + 1572864 + 49152 + 1572864;
    int*   idx          = (int*)(out + 1572864 + 49152 + 1572864 + 49152);
    int*   idx_new      = (int*)(out + 1572864 + 49152 + 1572864 + 49152 + 524288);

    fps_kernel<<<BATCH, FPS_THREADS, 0, stream>>>(xyz, nrm, idx_new, center, norm_center);

    const size_t lds = (size_t)CENT_PER_WG * NPTS * 2 + (CENT_PER_WG * 4) * sizeof(float);
    knn_group_kernel<<<BATCH * (NGRP / CENT_PER_WG), KNN_THREADS, lds, stream>>>(
        xyz, nrm, center, neighborhood, norm_grouped, idx);
}